// EfficientAttention_73375221284963
// MI455X (gfx1250) — compile-verified
//
#include <hip/hip_runtime.h>
#include <math.h>

typedef __attribute__((ext_vector_type(2))) float v2f;
typedef __attribute__((ext_vector_type(8))) float v8f;

// ---------------------------------------------------------------------------
// fp32 WMMA GEMM:  C[M,N] = A[M,K] * W[N,K]^T  (+ optional bias[N])
// Block = 256 threads (8 waves). Block tile 128x128, K-chunk 32.
// Double-buffered LDS + register prefetch: one barrier per K-chunk,
// next chunk's global loads in flight during WMMA compute.
// LDS row stride 36 floats: conflict-free ds_load_b64 fragments,
// 16B-aligned ds_store_b128 fills.
// ---------------------------------------------------------------------------
#define BM 128
#define BN 128
#define BK 32
#define SK 36   // LDS row stride in floats

__global__ __launch_bounds__(256) void gemm_wmma_f32(
    const float* __restrict__ A,   // [M,K] row-major
    const float* __restrict__ W,   // [N,K] row-major (used transposed)
    const float* __restrict__ bias,// [N] or nullptr
    float* __restrict__ C,         // [M,N]
    int M, int N, int K)
{
    __shared__ float lA[2][BM * SK];
    __shared__ float lW[2][BN * SK];

    const int t    = threadIdx.x;
    const int wave = t >> 5;
    const int lane = t & 31;
    const int mw   = wave >> 2;        // 0..1 -> M offset mw*64
    const int nw   = wave & 3;         // 0..3 -> N offset nw*32
    const int lr   = lane & 15;        // row within 16x16 tile
    const int kh   = (lane >> 4) << 1; // 0 or 2 : K-half per ISA A/B layout
    const int hi8  = (lane >> 4) << 3; // 0 or 8 : M-half for C/D layout

    const int m0 = blockIdx.y * BM;
    const int n0 = blockIdx.x * BN;

    // ---- staging geometry (invariant across chunks): thread t covers
    // 4 float4's of the A tile and 4 of the W tile per chunk.
    int srow[4], soff[4];                 // tile row, LDS float offset
    const float* pa[4];                   // rolling global pointers (advance BK)
    const float* pw[4];
#pragma unroll
    for (int u = 0; u < 4; ++u) {
        int idx  = t + u * 256;           // 0..1023
        int row  = idx >> 3;              // /8
        int c4   = (idx & 7) << 2;        // *4
        srow[u]  = row;
        soff[u]  = row * SK + c4;
        pa[u]    = A + (size_t)(m0 + row) * K + c4;
        pw[u]    = W + (size_t)(n0 + row) * K + c4;
    }

    v8f acc[4][2];
#pragma unroll
    for (int mt = 0; mt < 4; ++mt)
#pragma unroll
        for (int nt = 0; nt < 2; ++nt)
            acc[mt][nt] = (v8f){0.f,0.f,0.f,0.f,0.f,0.f,0.f,0.f};

    // ---- prologue: load chunk 0 into registers
    float4 ra[4], rw[4];
#pragma unroll
    for (int u = 0; u < 4; ++u) {
        ra[u] = *(const float4*)pa[u];  pa[u] += BK;
        rw[u] = *(const float4*)pw[u];  pw[u] += BK;
    }

    const int nc = K / BK;
    for (int c = 0; c < nc; ++c) {
        const int buf = c & 1;

        // commit current chunk's registers to LDS
#pragma unroll
        for (int u = 0; u < 4; ++u) {
            *(float4*)&lA[buf][soff[u]] = ra[u];
            *(float4*)&lW[buf][soff[u]] = rw[u];
        }

        // issue next chunk's global loads (consumed next iteration)
        if (c + 1 < nc) {
#pragma unroll
            for (int u = 0; u < 4; ++u) {
                ra[u] = *(const float4*)pa[u];  pa[u] += BK;
                rw[u] = *(const float4*)pw[u];  pw[u] += BK;
            }
        }

        __syncthreads();   // lds[buf] visible to all; also fences buf^1 reuse

        // ---- 8 k-steps of 4, 8 WMMAs each
#pragma unroll
        for (int k = 0; k < BK; k += 4) {
            v2f af[4], bf[2];
#pragma unroll
            for (int mt = 0; mt < 4; ++mt) {
                int row = mw * 64 + mt * 16 + lr;
                af[mt] = *(const v2f*)&lA[buf][row * SK + k + kh];
            }
#pragma unroll
            for (int nt = 0; nt < 2; ++nt) {
                int row = nw * 32 + nt * 16 + lr;
                bf[nt] = *(const v2f*)&lW[buf][row * SK + k + kh];
            }
#pragma unroll
            for (int mt = 0; mt < 4; ++mt)
#pragma unroll
                for (int nt = 0; nt < 2; ++nt)
                    acc[mt][nt] = __builtin_amdgcn_wmma_f32_16x16x4_f32(
                        false, af[mt], false, bf[nt],
                        (short)0, acc[mt][nt], false, false);
        }
    }

    // ---- store: VGPR rr -> (M = rr + hi8, N = lr) per C/D layout
#pragma unroll
    for (int mt = 0; mt < 4; ++mt) {
#pragma unroll
        for (int nt = 0; nt < 2; ++nt) {
            int col = n0 + nw * 32 + nt * 16 + lr;
            float bv = (bias != nullptr) ? bias[col] : 0.f;
#pragma unroll
            for (int rr = 0; rr < 8; ++rr) {
                int row = m0 + mw * 64 + mt * 16 + hi8 + rr;
                C[(size_t)row * N + col] = acc[mt][nt][rr] + bv;
            }
        }
    }
}

// ---------------------------------------------------------------------------
// Band attention, window 7 (|i-j| <= 3). One wave per (b, h, query i).
// qkv layout: [B, N, 3C] with q at +0, k at +C, v at +2C, head slice h*64.
// Lane l owns feature components l and l+32 of the d=64 head dim.
// ---------------------------------------------------------------------------
__global__ __launch_bounds__(256) void attn_band7(
    const float* __restrict__ qkv,   // [B, N, 3*C]
    const float* __restrict__ temp,  // [H]
    float* __restrict__ aout,        // [B, N, C]
    int B, int N, int H, int C)
{
    const int C3 = 3 * C;
    const int wid  = (blockIdx.x * blockDim.x + threadIdx.x) >> 5;
    const int lane = threadIdx.x & 31;
    const int i = wid % N;
    const int h = (wid / N) % H;
    const int b = wid / (N * H);

    const float* qp = qkv + (size_t)(b * N + i) * C3 + h * 64;
    const float q0 = qp[lane];
    const float q1 = qp[lane + 32];
    const float invt = 0.125f / temp[h];   // scale = d^-0.5 = 1/8, then /temperature

    float logits[7];
#pragma unroll
    for (int jj = 0; jj < 7; ++jj) {
        const int j = i + jj - 3;
        const bool valid = (j >= 0) && (j < N);
        float p = 0.f;
        if (valid) {
            const float* kp = qkv + (size_t)(b * N + j) * C3 + C + h * 64;
            p = q0 * kp[lane] + q1 * kp[lane + 32];
        }
#pragma unroll
        for (int off = 16; off >= 1; off >>= 1)
            p += __shfl_xor(p, off, 32);
        logits[jj] = valid ? p * invt : -INFINITY;
    }

    float m = logits[0];
#pragma unroll
    for (int jj = 1; jj < 7; ++jj) m = fmaxf(m, logits[jj]);

    float w[7];
    float s = 0.f;
#pragma unroll
    for (int jj = 0; jj < 7; ++jj) { w[jj] = __expf(logits[jj] - m); s += w[jj]; }
    const float inv = 1.f / s;

    float o0 = 0.f, o1 = 0.f;
#pragma unroll
    for (int jj = 0; jj < 7; ++jj) {
        const int j = i + jj - 3;
        if (j >= 0 && j < N) {
            const float* vp = qkv + (size_t)(b * N + j) * C3 + 2 * C + h * 64;
            o0 += w[jj] * vp[lane];
            o1 += w[jj] * vp[lane + 32];
        }
    }
    float* op = aout + (size_t)(b * N + i) * C + h * 64;
    op[lane]      = o0 * inv;
    op[lane + 32] = o1 * inv;
}

// ---------------------------------------------------------------------------
extern "C" void kernel_launch(void* const* d_in, const int* in_sizes, int n_in,
                              void* d_out, int out_size, void* d_ws, size_t ws_size,
                              hipStream_t stream)
{
    (void)in_sizes; (void)n_in; (void)out_size; (void)ws_size;

    const int B = 8, N = 1024, C = 768, H = 12;
    const int M  = B * N;        // 8192
    const int C3 = 3 * C;        // 2304

    const float* x      = (const float*)d_in[0];  // [B,N,C]
    const float* W_qkv  = (const float*)d_in[1];  // [3C,C]
    const float* W_proj = (const float*)d_in[2];  // [C,C]
    const float* b_proj = (const float*)d_in[3];  // [C]
    const float* temp   = (const float*)d_in[4];  // [H,1,1]
    float*       out    = (float*)d_out;          // [B,N,C]

    float* qkv  = (float*)d_ws;                   // [M, 3C]  ~75.5 MB
    float* aout = qkv + (size_t)M * C3;           // [M, C]   ~25.2 MB

    // 1) qkv = x @ W_qkv^T        (M=8192, N=2304, K=768)
    {
        dim3 grid(C3 / BN, M / BM);
        gemm_wmma_f32<<<grid, 256, 0, stream>>>(x, W_qkv, nullptr, qkv, M, C3, C);
    }

    // 2) band attention (window 7), one wave per (b,h,i)
    {
        int waves  = B * H * N;           // 98304
        int blocks = waves / 8;           // 8 waves / 256-thread block
        attn_band7<<<blocks, 256, 0, stream>>>(qkv, temp, aout, B, N, H, C);
    }

    // 3) out = aout @ W_proj^T + b_proj   (M=8192, N=768, K=768)
    {
        dim3 grid(C / BN, M / BM);
        gemm_wmma_f32<<<grid, 256, 0, stream>>>(aout, W_proj, b_proj, out, M, C, C);
    }
}